// NequIPConvolution_60189671686876
// MI455X (gfx1250) — compile-verified
//
#include <hip/hip_runtime.h>
#include <hip/hip_bf16.h>
#include <stdint.h>

// ---------------------------------------------------------------------------
// NequIP convolution for MI455X (gfx1250, wave32, WMMA).
// All GEMMs use V_WMMA_F32_16X16X4_F32; M=32 per wave so every B fragment
// feeds two WMMAs. Node-row staging uses GLOBAL_LOAD_ASYNC_TO_LDS_B128
// (ASYNCcnt path) with a plain-LDS fallback.
// ---------------------------------------------------------------------------

#define N_NODES 20000
#define N_EDGES 320000

typedef float v2f __attribute__((ext_vector_type(2)));
typedef float v8f __attribute__((ext_vector_type(8)));
typedef int   v4i __attribute__((ext_vector_type(4)));

#if defined(__has_builtin)
#if __has_builtin(__builtin_amdgcn_global_load_async_to_lds_b128)
#define HAVE_ASYNC_LDS 1
#endif
#endif
#ifndef HAVE_ASYNC_LDS
#define HAVE_ASYNC_LDS 0
#endif

typedef __attribute__((address_space(1))) v4i* gv4i_t;   // global 16B chunk
typedef __attribute__((address_space(3))) v4i* lv4i_t;   // LDS 16B chunk

__device__ __forceinline__ void async_wait0() {
#if __has_builtin(__builtin_amdgcn_s_wait_asynccnt)
    __builtin_amdgcn_s_wait_asynccnt(0);
#else
    asm volatile("s_wait_asynccnt 0x0" ::: "memory");
#endif
}

__device__ __forceinline__ void async_copy16(const float* g, float* l) {
#if HAVE_ASYNC_LDS
    __builtin_amdgcn_global_load_async_to_lds_b128(
        (gv4i_t)(uintptr_t)g, (lv4i_t)(uintptr_t)l, 0, 0);
#else
    *(float4*)l = *(const float4*)g;
#endif
}

// D = A(16x4 f32) * B(4x16 f32) + C(16x16 f32)
__device__ __forceinline__ v8f wmma4(v2f a, v2f b, v8f c) {
    return __builtin_amdgcn_wmma_f32_16x16x4_f32(
        /*neg_a=*/false, a, /*neg_b=*/false, b,
        /*c_mod=*/(short)0, c, /*reuse_a=*/false, /*reuse_b=*/false);
}

__device__ __forceinline__ float silu(float x) {
    return x / (1.0f + __expf(-x));
}

// ---------------------------------------------------------------------------
// Kernel 1: one wave = one 32-node tile.
//   xs  = x_s @ W_lin1_s * 1/8
//   xv  = x_v @ W_lin1_v * 1/8              (3 components)
//   scs = outer(x_s, attr) @ W_sc_s * 1/32  (K = 1024)
//   scv = outer(x_v, attr) @ W_sc_v * 1/32  (3 components, K = 1024)
// ---------------------------------------------------------------------------
__global__ __launch_bounds__(32) void node_stage_kernel(
    const float* __restrict__ nf,   // (N,256)
    const float* __restrict__ na,   // (N,16)
    const float* __restrict__ W1s,  // (64,64)
    const float* __restrict__ W1v,  // (64,64)
    const float* __restrict__ Wscs, // (1024,64)
    const float* __restrict__ Wscv, // (1024,64)
    float* __restrict__ xs,         // (N,64)
    float* __restrict__ xv,         // (N,64,3)
    float* __restrict__ scs,        // (N,64)
    float* __restrict__ scv)        // (N,64,3)
{
    __shared__ float xr[32][256];
    __shared__ float at[32][16];

    const int lane = threadIdx.x;
    const int m0   = blockIdx.x * 32;
    const int q    = lane & 15;
    const int hf   = lane >> 4;

    __builtin_prefetch(Wscs, 0, 2);
    __builtin_prefetch(Wscv, 0, 2);

    // Stage 32 node rows + attrs into LDS via the async-to-LDS path.
    for (int i = lane; i < 32 * 64; i += 32) {       // b128 chunks
        int n = i >> 6, c4 = i & 63;
        async_copy16(nf + (size_t)(m0 + n) * 256 + c4 * 4, &xr[n][c4 * 4]);
    }
    for (int i = lane; i < 32 * 4; i += 32) {
        int n = i >> 2, c4 = i & 3;
        async_copy16(na + (size_t)(m0 + n) * 16 + c4 * 4, &at[n][c4 * 4]);
    }
    async_wait0();
    __syncthreads();

    // ---- (a) xs = x_s @ W_lin1_s ----
    {
        v8f acc[2][4] = {};
        for (int k0 = 0; k0 < 64; k0 += 4) {
            const int kk = k0 + 2 * hf;
            v2f a0, a1;
            a0.x = xr[q][kk];      a0.y = xr[q][kk + 1];
            a1.x = xr[16 + q][kk]; a1.y = xr[16 + q][kk + 1];
#pragma unroll
            for (int nt = 0; nt < 4; ++nt) {
                const float* bp = W1s + kk * 64 + nt * 16 + q;
                v2f b; b.x = bp[0]; b.y = bp[64];
                acc[0][nt] = wmma4(a0, b, acc[0][nt]);
                acc[1][nt] = wmma4(a1, b, acc[1][nt]);
            }
        }
#pragma unroll
        for (int mt = 0; mt < 2; ++mt)
#pragma unroll
            for (int nt = 0; nt < 4; ++nt)
#pragma unroll
                for (int j = 0; j < 8; ++j)
                    xs[(size_t)(m0 + mt * 16 + j + 8 * hf) * 64 + nt * 16 + q] =
                        acc[mt][nt][j] * 0.125f;
    }

    // ---- (b) xv components ----
    for (int c = 0; c < 3; ++c) {
        v8f acc[2][4] = {};
        for (int k0 = 0; k0 < 64; k0 += 4) {
            const int kk = k0 + 2 * hf;
            v2f a0, a1;
            a0.x = xr[q][64 + kk * 3 + c];      a0.y = xr[q][64 + (kk + 1) * 3 + c];
            a1.x = xr[16 + q][64 + kk * 3 + c]; a1.y = xr[16 + q][64 + (kk + 1) * 3 + c];
#pragma unroll
            for (int nt = 0; nt < 4; ++nt) {
                const float* bp = W1v + kk * 64 + nt * 16 + q;
                v2f b; b.x = bp[0]; b.y = bp[64];
                acc[0][nt] = wmma4(a0, b, acc[0][nt]);
                acc[1][nt] = wmma4(a1, b, acc[1][nt]);
            }
        }
#pragma unroll
        for (int mt = 0; mt < 2; ++mt)
#pragma unroll
            for (int nt = 0; nt < 4; ++nt)
#pragma unroll
                for (int j = 0; j < 8; ++j)
                    xv[(size_t)(m0 + mt * 16 + j + 8 * hf) * 192 +
                       (nt * 16 + q) * 3 + c] = acc[mt][nt][j] * 0.125f;
    }

    // ---- (c) sc_s : K = 1024, A formed on the fly from LDS ----
    {
        v8f acc[2][4] = {};
        for (int k0 = 0; k0 < 1024; k0 += 4) {
            const int kk0 = k0 + 2 * hf, kk1 = kk0 + 1;
            const int u0 = kk0 >> 4, a0i = kk0 & 15;
            const int u1 = kk1 >> 4, a1i = kk1 & 15;
            v2f a0, a1;
            a0.x = xr[q][u0] * at[q][a0i];
            a0.y = xr[q][u1] * at[q][a1i];
            a1.x = xr[16 + q][u0] * at[16 + q][a0i];
            a1.y = xr[16 + q][u1] * at[16 + q][a1i];
#pragma unroll
            for (int nt = 0; nt < 4; ++nt) {
                const float* bp = Wscs + (size_t)kk0 * 64 + nt * 16 + q;
                v2f b; b.x = bp[0]; b.y = bp[64];
                acc[0][nt] = wmma4(a0, b, acc[0][nt]);
                acc[1][nt] = wmma4(a1, b, acc[1][nt]);
            }
        }
#pragma unroll
        for (int mt = 0; mt < 2; ++mt)
#pragma unroll
            for (int nt = 0; nt < 4; ++nt)
#pragma unroll
                for (int j = 0; j < 8; ++j)
                    scs[(size_t)(m0 + mt * 16 + j + 8 * hf) * 64 + nt * 16 + q] =
                        acc[mt][nt][j] * 0.03125f;
    }

    // ---- (d) sc_v components ----
    for (int c = 0; c < 3; ++c) {
        v8f acc[2][4] = {};
        for (int k0 = 0; k0 < 1024; k0 += 4) {
            const int kk0 = k0 + 2 * hf, kk1 = kk0 + 1;
            const int u0 = kk0 >> 4, a0i = kk0 & 15;
            const int u1 = kk1 >> 4, a1i = kk1 & 15;
            v2f a0, a1;
            a0.x = xr[q][64 + u0 * 3 + c] * at[q][a0i];
            a0.y = xr[q][64 + u1 * 3 + c] * at[q][a1i];
            a1.x = xr[16 + q][64 + u0 * 3 + c] * at[16 + q][a0i];
            a1.y = xr[16 + q][64 + u1 * 3 + c] * at[16 + q][a1i];
#pragma unroll
            for (int nt = 0; nt < 4; ++nt) {
                const float* bp = Wscv + (size_t)kk0 * 64 + nt * 16 + q;
                v2f b; b.x = bp[0]; b.y = bp[64];
                acc[0][nt] = wmma4(a0, b, acc[0][nt]);
                acc[1][nt] = wmma4(a1, b, acc[1][nt]);
            }
        }
#pragma unroll
        for (int mt = 0; mt < 2; ++mt)
#pragma unroll
            for (int nt = 0; nt < 4; ++nt)
#pragma unroll
                for (int j = 0; j < 8; ++j)
                    scv[(size_t)(m0 + mt * 16 + j + 8 * hf) * 192 +
                        (nt * 16 + q) * 3 + c] = acc[mt][nt][j] * 0.03125f;
    }
}

// ---------------------------------------------------------------------------
// Kernel 2: one wave = 32 edges. Fused MLP (8->64->64->256, WMMA + SiLU,
// LDS transposes) + tensor product + atomic scatter into agg[dst].
// ---------------------------------------------------------------------------
__global__ __launch_bounds__(32) void edge_fused_kernel(
    const int*   __restrict__ esrc,
    const int*   __restrict__ edst,
    const float* __restrict__ eattr,  // (E,4)
    const float* __restrict__ eemb,   // (E,8)
    const float* __restrict__ W0,     // (8,64)
    const float* __restrict__ W1,     // (64,64)
    const float* __restrict__ W2,     // (64,256)
    const float* __restrict__ xs,     // (N,64)
    const float* __restrict__ xv,     // (N,64,3)
    float* __restrict__ agg)          // (N,512)
{
    __shared__ float wt[32][256];
    __shared__ float hs[32][64];

    const int lane = threadIdx.x;
    const int e0   = blockIdx.x * 32;
    const int q    = lane & 15;
    const int hf   = lane >> 4;

    __builtin_prefetch(W2, 0, 2);

    // ---- GEMM0: h0 = silu(emb @ W_mlp0), K = 8 ----
    {
        v8f acc[2][4] = {};
        for (int k0 = 0; k0 < 8; k0 += 4) {
            const int kk = k0 + 2 * hf;
            const float* a0p = eemb + (size_t)(e0 + q) * 8 + kk;
            const float* a1p = eemb + (size_t)(e0 + 16 + q) * 8 + kk;
            v2f a0, a1;
            a0.x = a0p[0]; a0.y = a0p[1];
            a1.x = a1p[0]; a1.y = a1p[1];
#pragma unroll
            for (int nt = 0; nt < 4; ++nt) {
                const float* bp = W0 + kk * 64 + nt * 16 + q;
                v2f b; b.x = bp[0]; b.y = bp[64];
                acc[0][nt] = wmma4(a0, b, acc[0][nt]);
                acc[1][nt] = wmma4(a1, b, acc[1][nt]);
            }
        }
#pragma unroll
        for (int mt = 0; mt < 2; ++mt)
#pragma unroll
            for (int nt = 0; nt < 4; ++nt)
#pragma unroll
                for (int j = 0; j < 8; ++j)
                    hs[mt * 16 + j + 8 * hf][nt * 16 + q] = silu(acc[mt][nt][j]);
    }
    __syncthreads();

    // ---- GEMM1: h1 = silu(h0 @ W_mlp1), K = 64 ----
    {
        v8f acc[2][4] = {};
        for (int k0 = 0; k0 < 64; k0 += 4) {
            const int kk = k0 + 2 * hf;
            v2f a0, a1;
            a0.x = hs[q][kk];      a0.y = hs[q][kk + 1];
            a1.x = hs[16 + q][kk]; a1.y = hs[16 + q][kk + 1];
#pragma unroll
            for (int nt = 0; nt < 4; ++nt) {
                const float* bp = W1 + kk * 64 + nt * 16 + q;
                v2f b; b.x = bp[0]; b.y = bp[64];
                acc[0][nt] = wmma4(a0, b, acc[0][nt]);
                acc[1][nt] = wmma4(a1, b, acc[1][nt]);
            }
        }
        __syncthreads();   // all hs reads complete before overwrite
#pragma unroll
        for (int mt = 0; mt < 2; ++mt)
#pragma unroll
            for (int nt = 0; nt < 4; ++nt)
#pragma unroll
                for (int j = 0; j < 8; ++j)
                    hs[mt * 16 + j + 8 * hf][nt * 16 + q] = silu(acc[mt][nt][j]);
    }
    __syncthreads();

    // ---- GEMM2: w = h1 @ W_mlp2, K = 64, N = 256 (16 n-tiles) ----
    for (int nt = 0; nt < 16; ++nt) {
        v8f acc0 = {}, acc1 = {};
        for (int k0 = 0; k0 < 64; k0 += 4) {
            const int kk = k0 + 2 * hf;
            v2f a0, a1;
            a0.x = hs[q][kk];      a0.y = hs[q][kk + 1];
            a1.x = hs[16 + q][kk]; a1.y = hs[16 + q][kk + 1];
            const float* bp = W2 + (size_t)kk * 256 + nt * 16 + q;
            v2f b; b.x = bp[0]; b.y = bp[256];
            acc0 = wmma4(a0, b, acc0);
            acc1 = wmma4(a1, b, acc1);
        }
#pragma unroll
        for (int j = 0; j < 8; ++j) {
            wt[j + 8 * hf][nt * 16 + q]      = acc0[j];
            wt[16 + j + 8 * hf][nt * 16 + q] = acc1[j];
        }
    }
    __syncthreads();

    // ---- Tensor product + scatter: 32 edges x 64 channels over 32 lanes ----
    const float inv_s3 = 0.57735026918962576f;
    for (int it = 0; it < 64; ++it) {
        const int flat = it * 32 + lane;
        const int e    = flat >> 6;      // 0..31
        const int u    = flat & 63;      // 0..63
        const int edge = e0 + e;

        const int s = esrc[edge];
        const int d = edst[edge];
        const float ea0 = eattr[(size_t)edge * 4 + 0];
        const float ea1 = eattr[(size_t)edge * 4 + 1];
        const float ea2 = eattr[(size_t)edge * 4 + 2];
        const float ea3 = eattr[(size_t)edge * 4 + 3];

        const float xse = xs[(size_t)s * 64 + u];
        const float v0  = xv[(size_t)s * 192 + u * 3 + 0];
        const float v1  = xv[(size_t)s * 192 + u * 3 + 1];
        const float v2  = xv[(size_t)s * 192 + u * 3 + 2];

        const float w0 = wt[e][u];
        const float w1 = wt[e][64 + u];
        const float w2 = wt[e][128 + u];
        const float w3 = wt[e][192 + u];

        float* ag = agg + (size_t)d * 512;
        atomicAdd(ag + u, w0 * xse * ea0);                       // out0
        const float dot = v0 * ea1 + v1 * ea2 + v2 * ea3;
        atomicAdd(ag + 64 + u, w3 * dot * inv_s3);               // out3
        const float t1 = w1 * xse;                               // out1
        atomicAdd(ag + 128 + u * 3 + 0, t1 * ea1);
        atomicAdd(ag + 128 + u * 3 + 1, t1 * ea2);
        atomicAdd(ag + 128 + u * 3 + 2, t1 * ea3);
        const float t2 = w2 * ea0;                               // out2
        atomicAdd(ag + 320 + u * 3 + 0, t2 * v0);
        atomicAdd(ag + 320 + u * 3 + 1, t2 * v1);
        atomicAdd(ag + 320 + u * 3 + 2, t2 * v2);
    }
}

// ---------------------------------------------------------------------------
// Kernel 3: output linears (K = 128 WMMA GEMMs on agg) + skip connection.
// One wave = 32 nodes.
// ---------------------------------------------------------------------------
__global__ __launch_bounds__(32) void node_out_kernel(
    const float* __restrict__ agg,  // (N,512)
    const float* __restrict__ W2s,  // (128,64)
    const float* __restrict__ W2v,  // (128,64)
    const float* __restrict__ scs,  // (N,64)
    const float* __restrict__ scv,  // (N,64,3)
    float* __restrict__ out)        // (N,256)
{
    const int lane = threadIdx.x;
    const int m0   = blockIdx.x * 32;
    const int q    = lane & 15;
    const int hf   = lane >> 4;
    const float scale = 0.25f * 0.088388347648318447f;  // inv_nb * inv_2m

    // ---- y_s ----
    {
        v8f acc[2][4] = {};
        for (int k0 = 0; k0 < 128; k0 += 4) {
            const int kk = k0 + 2 * hf;
            const float* a0p = agg + (size_t)(m0 + q) * 512 + kk;
            const float* a1p = agg + (size_t)(m0 + 16 + q) * 512 + kk;
            v2f a0, a1;
            a0.x = a0p[0]; a0.y = a0p[1];
            a1.x = a1p[0]; a1.y = a1p[1];
#pragma unroll
            for (int nt = 0; nt < 4; ++nt) {
                const float* bp = W2s + kk * 64 + nt * 16 + q;
                v2f b; b.x = bp[0]; b.y = bp[64];
                acc[0][nt] = wmma4(a0, b, acc[0][nt]);
                acc[1][nt] = wmma4(a1, b, acc[1][nt]);
            }
        }
#pragma unroll
        for (int mt = 0; mt < 2; ++mt)
#pragma unroll
            for (int nt = 0; nt < 4; ++nt)
#pragma unroll
                for (int j = 0; j < 8; ++j) {
                    const int row = m0 + mt * 16 + j + 8 * hf;
                    const int col = nt * 16 + q;
                    out[(size_t)row * 256 + col] =
                        acc[mt][nt][j] * scale + scs[(size_t)row * 64 + col];
                }
    }

    // ---- y_v components ----
    for (int c = 0; c < 3; ++c) {
        v8f acc[2][4] = {};
        for (int k0 = 0; k0 < 128; k0 += 4) {
            const int kk = k0 + 2 * hf;
            const float* a0p = agg + (size_t)(m0 + q) * 512 + 128;
            const float* a1p = agg + (size_t)(m0 + 16 + q) * 512 + 128;
            v2f a0, a1;
            a0.x = a0p[kk * 3 + c];       a0.y = a0p[(kk + 1) * 3 + c];
            a1.x = a1p[kk * 3 + c];       a1.y = a1p[(kk + 1) * 3 + c];
#pragma unroll
            for (int nt = 0; nt < 4; ++nt) {
                const float* bp = W2v + kk * 64 + nt * 16 + q;
                v2f b; b.x = bp[0]; b.y = bp[64];
                acc[0][nt] = wmma4(a0, b, acc[0][nt]);
                acc[1][nt] = wmma4(a1, b, acc[1][nt]);
            }
        }
#pragma unroll
        for (int mt = 0; mt < 2; ++mt)
#pragma unroll
            for (int nt = 0; nt < 4; ++nt)
#pragma unroll
                for (int j = 0; j < 8; ++j) {
                    const int row = m0 + mt * 16 + j + 8 * hf;
                    const int col = nt * 16 + q;
                    out[(size_t)row * 256 + 64 + col * 3 + c] =
                        acc[mt][nt][j] * scale + scv[(size_t)row * 192 + col * 3 + c];
                }
    }
}

// ---------------------------------------------------------------------------
extern "C" void kernel_launch(void* const* d_in, const int* in_sizes, int n_in,
                              void* d_out, int out_size, void* d_ws, size_t ws_size,
                              hipStream_t stream) {
    const float* nf    = (const float*)d_in[0];
    const float* na    = (const float*)d_in[1];
    const int*   esrc  = (const int*)  d_in[2];
    const int*   edst  = (const int*)  d_in[3];
    const float* eattr = (const float*)d_in[4];
    const float* eemb  = (const float*)d_in[5];
    const float* W1s   = (const float*)d_in[6];
    const float* W1v   = (const float*)d_in[7];
    const float* Wm0   = (const float*)d_in[8];
    const float* Wm1   = (const float*)d_in[9];
    const float* Wm2   = (const float*)d_in[10];
    const float* W2s   = (const float*)d_in[11];
    const float* W2v   = (const float*)d_in[12];
    const float* Wscs  = (const float*)d_in[13];
    const float* Wscv  = (const float*)d_in[14];

    float* ws  = (float*)d_ws;
    float* xs  = ws;                         // N*64
    float* xv  = ws + (size_t)N_NODES * 64;  // N*192
    float* scs = ws + (size_t)N_NODES * 256; // N*64
    float* scv = ws + (size_t)N_NODES * 320; // N*192
    float* agg = ws + (size_t)N_NODES * 512; // N*512
    float* out = (float*)d_out;

    (void)hipMemsetAsync(agg, 0, (size_t)N_NODES * 512 * sizeof(float), stream);

    node_stage_kernel<<<N_NODES / 32, 32, 0, stream>>>(
        nf, na, W1s, W1v, Wscs, Wscv, xs, xv, scs, scv);

    edge_fused_kernel<<<N_EDGES / 32, 32, 0, stream>>>(
        esrc, edst, eattr, eemb, Wm0, Wm1, Wm2, xs, xv, agg);

    node_out_kernel<<<N_NODES / 32, 32, 0, stream>>>(
        agg, W2s, W2v, scs, scv, out);
}